// ResNetTTT_62989990363333
// MI455X (gfx1250) — compile-verified
//
#include <hip/hip_runtime.h>
#include <math.h>

// ---------------------------------------------------------------------------
// Types for CDNA5 fp32 WMMA: V_WMMA_F32_16X16X4_F32
//   A: 16x4  -> 2 VGPRs (lane=M within half, vgpr/half = K slot)
//   B: 4x16  -> 2 VGPRs (lane=N within half, vgpr/half = K slot)
//   C/D: 16x16 -> 8 VGPRs (vgpr i: M=i / M=8+i per lane half, N=lane&15)
// ---------------------------------------------------------------------------
typedef __attribute__((ext_vector_type(2))) float v2f;
typedef __attribute__((ext_vector_type(8))) float v8f;

__device__ __forceinline__ v8f wmma4(v2f a, v2f b, v8f c) {
  // 8 args: (neg_a, A, neg_b, B, c_mod, C, reuse_a, reuse_b)
  return __builtin_amdgcn_wmma_f32_16x16x4_f32(false, a, false, b, (short)0, c,
                                               false, false);
}

__device__ __forceinline__ float geluf(float z) {
  return 0.5f * z * (1.0f + erff(z * 0.7071067811865476f));
}
__device__ __forceinline__ float gelu_gradf(float z) {
  return 0.5f * (1.0f + erff(z * 0.7071067811865476f)) +
         z * expf(-0.5f * z * z) * 0.3989422804014327f;
}

// Optional gfx1250 async Global->LDS path (hedged: only if builtins exist).
// Probe round 2: builtin exists; param 0 is (address_space(1)) int*.
#if defined(__has_builtin)
#if __has_builtin(__builtin_amdgcn_global_load_async_to_lds_b32) && \
    __has_builtin(__builtin_amdgcn_s_wait_asynccnt)
#define USE_ASYNC_LDS 1
#endif
#endif

#if defined(USE_ASYNC_LDS)
typedef __attribute__((address_space(1))) int as1_i32;
typedef __attribute__((address_space(3))) int as3_i32;
__device__ __forceinline__ void async_copy_b32(const float* g, float* lds) {
  __builtin_amdgcn_global_load_async_to_lds_b32((as1_i32*)g, (as3_i32*)lds, 0,
                                                0);
}
#endif

// ---------------------------------------------------------------------------
// Generic fp32 WMMA GEMM: C[M,N] = act(A[M,K] @ W + bias + res)
// TRANSB: W stored [N][K]; else [K][N]. One wave computes a 16x64 strip
// (4 accumulators), reusing the A fragment across 4 B fragments.
// Requires N % 64 == 0, M % 16 == 0, K % 4 == 0.
// ---------------------------------------------------------------------------
template <bool TRANSB, bool BIAS, bool RELU, bool RES>
__global__ void __launch_bounds__(256) gemm_wmma(
    const float* __restrict__ A, const float* __restrict__ W,
    const float* __restrict__ bias, const float* __restrict__ res,
    float* __restrict__ C, int M, int N, int K) {
  int lane = threadIdx.x & 31;
  int wave = threadIdx.x >> 5;
  int colGroups = N >> 6;
  int unit = blockIdx.x * 8 + wave;
  if (unit >= (M >> 4) * colGroups) return;  // wave-uniform
  int tm = (unit / colGroups) << 4;
  int tn0 = (unit % colGroups) << 6;
  int half = lane >> 4;
  int l15 = lane & 15;
  int arow = tm + l15;
  const v2f* arowp = (const v2f*)(A + (size_t)arow * K);
  v8f acc0 = {0.f, 0.f, 0.f, 0.f, 0.f, 0.f, 0.f, 0.f};
  v8f acc1 = acc0, acc2 = acc0, acc3 = acc0;
  for (int k = 0; k < K; k += 4) {
    __builtin_prefetch(&A[(size_t)arow * K + k + 32], 0, 0);
    int ka = k + half * 2;
    v2f a = arowp[ka >> 1];
    v2f b0, b1, b2, b3;
    if (TRANSB) {
      b0 = *(const v2f*)(W + (size_t)(tn0 + l15) * K + ka);
      b1 = *(const v2f*)(W + (size_t)(tn0 + 16 + l15) * K + ka);
      b2 = *(const v2f*)(W + (size_t)(tn0 + 32 + l15) * K + ka);
      b3 = *(const v2f*)(W + (size_t)(tn0 + 48 + l15) * K + ka);
    } else {
      const float* w0 = W + (size_t)ka * N + tn0 + l15;
      const float* w1 = W + (size_t)(ka + 1) * N + tn0 + l15;
      b0.x = w0[0];  b0.y = w1[0];
      b1.x = w0[16]; b1.y = w1[16];
      b2.x = w0[32]; b2.y = w1[32];
      b3.x = w0[48]; b3.y = w1[48];
    }
    acc0 = wmma4(a, b0, acc0);
    acc1 = wmma4(a, b1, acc1);
    acc2 = wmma4(a, b2, acc2);
    acc3 = wmma4(a, b3, acc3);
  }
  int crow0 = tm + half * 8;
  v8f accs[4] = {acc0, acc1, acc2, acc3};
  for (int j = 0; j < 4; j++) {
    int col = tn0 + j * 16 + l15;
    for (int i = 0; i < 8; i++) {
      float v = accs[j][i];
      int r = crow0 + i;
      if (BIAS) v += bias[col];
      if (RES) v += res[(size_t)r * N + col];
      if (RELU) v = fmaxf(v, 0.f);
      C[(size_t)r * N + col] = v;
    }
  }
}

// ---------------------------------------------------------------------------
// 3-tap conv over sequence dim as 3 shifted WMMA GEMMs. h: (B*L, 256).
// weights: (co, ci, 3). sumMode: 0 none, 1 store, 2 accumulate (skip sum).
// Branchless boundary handling: clamped row address * 0/1 mask.
// One wave computes a 16x64 strip.
// ---------------------------------------------------------------------------
__global__ void __launch_bounds__(256) conv3_wmma(
    const float* __restrict__ h, const float* __restrict__ w,
    const float* __restrict__ bias, float* __restrict__ out,
    float* __restrict__ sum, int sumMode) {
  const int Cc = 256;
  int lane = threadIdx.x & 31;
  int wave = threadIdx.x >> 5;
  int unit = blockIdx.x * 8 + wave;  // 256 row tiles * 4 col groups = 1024
  int tm = (unit >> 2) << 4;
  int tn0 = (unit & 3) << 6;
  int half = lane >> 4;
  int l15 = lane & 15;
  int r = tm + l15;           // output row for this lane's A rows
  int rlo = (r >> 10) << 10;  // batch segment [rlo, rlo+1024)
  int rhi = rlo + 1024;
  v8f acc0 = {0.f, 0.f, 0.f, 0.f, 0.f, 0.f, 0.f, 0.f};
  v8f acc1 = acc0, acc2 = acc0, acc3 = acc0;
  for (int t = 0; t < 3; t++) {
    int ar = r + t - 1;
    float okf = (ar >= rlo && ar < rhi) ? 1.f : 0.f;
    int arc = ar < rlo ? rlo : (ar >= rhi ? rhi - 1 : ar);
    const v2f* ap = (const v2f*)(h + (size_t)arc * Cc);
    for (int k = 0; k < Cc; k += 4) {
      int ka = k + half * 2;
      v2f a = ap[ka >> 1];
      a.x *= okf;
      a.y *= okf;
      v2f b0, b1, b2, b3;
      {
        const float* wp = w + (size_t)ka * 3 + t;  // + co*Cc*3
        size_t cstep = (size_t)Cc * 3;
        const float* c0 = wp + (size_t)(tn0 + l15) * cstep;
        const float* c1 = wp + (size_t)(tn0 + 16 + l15) * cstep;
        const float* c2 = wp + (size_t)(tn0 + 32 + l15) * cstep;
        const float* c3 = wp + (size_t)(tn0 + 48 + l15) * cstep;
        b0.x = c0[0]; b0.y = c0[3];
        b1.x = c1[0]; b1.y = c1[3];
        b2.x = c2[0]; b2.y = c2[3];
        b3.x = c3[0]; b3.y = c3[3];
      }
      acc0 = wmma4(a, b0, acc0);
      acc1 = wmma4(a, b1, acc1);
      acc2 = wmma4(a, b2, acc2);
      acc3 = wmma4(a, b3, acc3);
    }
  }
  int crow0 = tm + half * 8;
  v8f accs[4] = {acc0, acc1, acc2, acc3};
  for (int j = 0; j < 4; j++) {
    int co = tn0 + j * 16 + l15;
    for (int i = 0; i < 8; i++) {
      int rr = crow0 + i;
      float v = fmaxf(accs[j][i] + bias[co], 0.f);
      out[(size_t)rr * Cc + co] = v;
      if (sumMode == 1) sum[(size_t)rr * Cc + co] = v;
      else if (sumMode == 2) sum[(size_t)rr * Cc + co] += v;
    }
  }
}

// ---------------------------------------------------------------------------
// LayerNorm over C=256, one block per row.
// ---------------------------------------------------------------------------
__global__ void __launch_bounds__(256) layernorm256(
    const float* __restrict__ x, const float* __restrict__ w,
    const float* __restrict__ b, float* __restrict__ y) {
  __shared__ float s1[256], s2[256];
  int r = blockIdx.x;
  int t = threadIdx.x;
  float v = x[(size_t)r * 256 + t];
  s1[t] = v;
  s2[t] = v * v;
  __syncthreads();
  for (int off = 128; off > 0; off >>= 1) {
    if (t < off) {
      s1[t] += s1[t + off];
      s2[t] += s2[t + off];
    }
    __syncthreads();
  }
  float mu = s1[0] * (1.f / 256.f);
  float var = s2[0] * (1.f / 256.f) - mu * mu;
  float inv = rsqrtf(var + 1e-5f);
  y[(size_t)r * 256 + t] = w[t] * (v - mu) * inv + b[t];
}

// ---------------------------------------------------------------------------
// TTT scan: one workgroup per (b,h) chain; fast weights resident in LDS.
// W1 padded stride 257, W2 padded stride 65 (bank-conflict-free).
// Rank-4 weight updates done with V_WMMA_F32_16X16X4_F32 (K = MB = 4).
// Per-step activations fetched via async Global->LDS when available.
// ---------------------------------------------------------------------------
#define S1 257
#define S2 65

__global__ void __launch_bounds__(256) ttt_scan(
    const float* __restrict__ q, const float* __restrict__ k,
    const float* __restrict__ v, const float* __restrict__ W1g,
    const float* __restrict__ b1g, const float* __restrict__ W2g,
    const float* __restrict__ b2g, const float* __restrict__ lnw_g,
    const float* __restrict__ lnb_g, float* __restrict__ out) {
  extern __shared__ float sm[];
  float* W1s = sm;               // 64*S1
  float* W2s = W1s + 64 * S1;    // 256*S2
  float* b1s = W2s + 256 * S2;   // 256
  float* b2s = b1s + 256;        // 64
  float* lnw = b2s + 64;         // 64
  float* lnb = lnw + 64;         // 64
  float* xqs = lnb + 64;         // 4x64
  float* xks = xqs + 256;        // 4x64
  float* tgt = xks + 256;        // 4x64
  float* Z1s = tgt + 256;        // 4x256
  float* X2s = Z1s + 1024;       // 4x256
  float* G1s = X2s + 1024;       // 4x256
  float* Z2s = G1s + 1024;       // 4x64
  float* G2s = Z2s + 256;        // 4x64
  float* ZHs = G2s + 256;        // 4x64
  float* red = ZHs + 256;        // 4 rows x {mu, std, gmean, gzmean}

  const int tid = threadIdx.x;
  const int lane = tid & 31;
  const int wave = tid >> 5;
  const int l15 = lane & 15;
  const int half = lane >> 4;
  const int bh = blockIdx.x;  // 16 chains
  const int bb = bh >> 2;
  const int hh = bh & 3;
  const int m4 = tid >> 6;   // minibatch row 0..3
  const int n64 = tid & 63;  // feature 0..63
  const float eta = 0.1f / 4.0f;

  for (int i = tid; i < 64 * 256; i += 256)
    W1s[(i >> 8) * S1 + (i & 255)] = W1g[hh * 16384 + i];
  for (int i = tid; i < 256 * 64; i += 256)
    W2s[(i >> 6) * S2 + (i & 63)] = W2g[hh * 16384 + i];
  b1s[tid] = b1g[hh * 256 + tid];
  if (tid < 64) {
    b2s[tid] = b2g[hh * 64 + tid];
    lnw[tid] = lnw_g[hh * 64 + tid];
    lnb[tid] = lnb_g[hh * 64 + tid];
  }
  __syncthreads();

  for (int mb = 0; mb < 256; mb++) {
    const int l = mb * 4 + m4;
    const int gro = (bb * 1024 + l) * 256 + hh * 64 + n64;
    {
      float xvv = v[gro];
#if defined(USE_ASYNC_LDS)
      async_copy_b32(k + gro, &xks[tid]);
      async_copy_b32(q + gro, &xqs[tid]);
      __builtin_amdgcn_s_wait_asynccnt(0);
      tgt[tid] = xvv - xks[tid];
#else
      float xkv = k[gro];
      xks[tid] = xkv;
      xqs[tid] = q[gro];
      tgt[tid] = xvv - xkv;
#endif
    }
    __syncthreads();

    // Z1 = xk @ W1 + b1; X2 = gelu(Z1). Thread owns column tid, rows 0..3.
    {
      float z0 = b1s[tid], z1 = z0, z2 = z0, z3 = z0;
      for (int dd = 0; dd < 64; dd++) {
        float wv = W1s[dd * S1 + tid];
        z0 += xks[dd] * wv;
        z1 += xks[64 + dd] * wv;
        z2 += xks[128 + dd] * wv;
        z3 += xks[192 + dd] * wv;
      }
      Z1s[tid] = z0;        X2s[tid] = geluf(z0);
      Z1s[256 + tid] = z1;  X2s[256 + tid] = geluf(z1);
      Z1s[512 + tid] = z2;  X2s[512 + tid] = geluf(z2);
      Z1s[768 + tid] = z3;  X2s[768 + tid] = geluf(z3);
    }
    __syncthreads();
    // Z2 = X2 @ W2 + b2. Thread owns (m4, n64).
    {
      float z = b2s[n64];
      for (int kk = 0; kk < 256; kk++) z += X2s[m4 * 256 + kk] * W2s[kk * S2 + n64];
      Z2s[tid] = z;
    }
    __syncthreads();
    // ln_l2_bwd(Z2, lnw, lnb, tgt) -> G2s
    if (tid < 4) {
      float s = 0.f, ss = 0.f;
      for (int n = 0; n < 64; n++) {
        float z = Z2s[tid * 64 + n];
        s += z; ss += z * z;
      }
      float mu = s * (1.f / 64.f);
      red[tid * 4 + 0] = mu;
      red[tid * 4 + 1] = sqrtf(ss * (1.f / 64.f) - mu * mu + 1e-6f);
    }
    __syncthreads();
    {
      float mu = red[m4 * 4 + 0], sd = red[m4 * 4 + 1];
      float zh = (Z2s[tid] - mu) / sd;
      float y = lnw[n64] * zh + lnb[n64];
      ZHs[tid] = zh;
      G2s[tid] = lnw[n64] * (y - tgt[tid]);
    }
    __syncthreads();
    if (tid < 4) {
      float sg = 0.f, sgz = 0.f;
      for (int n = 0; n < 64; n++) {
        float g = G2s[tid * 64 + n];
        sg += g; sgz += g * ZHs[tid * 64 + n];
      }
      red[tid * 4 + 2] = sg * (1.f / 64.f);
      red[tid * 4 + 3] = sgz * (1.f / 64.f);
    }
    __syncthreads();
    {
      G2s[tid] = (G2s[tid] - red[m4 * 4 + 2] - ZHs[tid] * red[m4 * 4 + 3]) /
                 red[m4 * 4 + 1];
    }
    __syncthreads();
    // gZ1 = (gZ2 @ W2^T) * gelu'(Z1). Thread owns column tid, rows 0..3.
    {
      float g0 = 0.f, g1 = 0.f, g2 = 0.f, g3 = 0.f;
      for (int dd = 0; dd < 64; dd++) {
        float wv = W2s[tid * S2 + dd];
        g0 += G2s[dd] * wv;
        g1 += G2s[64 + dd] * wv;
        g2 += G2s[128 + dd] * wv;
        g3 += G2s[192 + dd] * wv;
      }
      G1s[tid] = g0 * gelu_gradf(Z1s[tid]);
      G1s[256 + tid] = g1 * gelu_gradf(Z1s[256 + tid]);
      G1s[512 + tid] = g2 * gelu_gradf(Z1s[512 + tid]);
      G1s[768 + tid] = g3 * gelu_gradf(Z1s[768 + tid]);
    }
    __syncthreads();
    // Bias updates.
    b1s[tid] -= eta * (G1s[tid] + G1s[256 + tid] + G1s[512 + tid] + G1s[768 + tid]);
    if (tid < 64)
      b2s[tid] -= eta * (G2s[tid] + G2s[64 + tid] + G2s[128 + tid] + G2s[192 + tid]);
    // W1 -= eta * xk^T @ gZ1 : 64 tiles of 16x16, K=4 -> one WMMA each.
    for (int tIdx = wave; tIdx < 64; tIdx += 8) {
      int tr = tIdx >> 4;   // 4 row blocks of W1 (64 rows)
      int tc = tIdx & 15;   // 16 col tiles (256 cols)
      int mm = tr * 16 + l15;
      v2f a, bf;
      a.x = -eta * xks[(half * 2) * 64 + mm];
      a.y = -eta * xks[(half * 2 + 1) * 64 + mm];
      bf.x = G1s[(half * 2) * 256 + tc * 16 + l15];
      bf.y = G1s[(half * 2 + 1) * 256 + tc * 16 + l15];
      int baseRow = tr * 16 + half * 8;
      int col = tc * 16 + l15;
      v8f c;
      for (int i = 0; i < 8; i++) c[i] = W1s[(baseRow + i) * S1 + col];
      c = wmma4(a, bf, c);
      for (int i = 0; i < 8; i++) W1s[(baseRow + i) * S1 + col] = c[i];
    }
    // W2 -= eta * X2^T @ gZ2 : 64 tiles of 16x16.
    for (int tIdx = wave; tIdx < 64; tIdx += 8) {
      int tr = tIdx >> 2;  // 16 row blocks (256 rows)
      int tc = tIdx & 3;   // 4 col tiles (64 cols)
      int mm = tr * 16 + l15;
      v2f a, bf;
      a.x = -eta * X2s[(half * 2) * 256 + mm];
      a.y = -eta * X2s[(half * 2 + 1) * 256 + mm];
      bf.x = G2s[(half * 2) * 64 + tc * 16 + l15];
      bf.y = G2s[(half * 2 + 1) * 64 + tc * 16 + l15];
      int baseRow = tr * 16 + half * 8;
      int col = tc * 16 + l15;
      v8f c;
      for (int i = 0; i < 8; i++) c[i] = W2s[(baseRow + i) * S2 + col];
      c = wmma4(a, bf, c);
      for (int i = 0; i < 8; i++) W2s[(baseRow + i) * S2 + col] = c[i];
    }
    __syncthreads();
    // Z1q / Z2q with updated weights, then out = xq + ln(Z2q).
    {
      float z0 = b1s[tid], z1 = z0, z2 = z0, z3 = z0;
      for (int dd = 0; dd < 64; dd++) {
        float wv = W1s[dd * S1 + tid];
        z0 += xqs[dd] * wv;
        z1 += xqs[64 + dd] * wv;
        z2 += xqs[128 + dd] * wv;
        z3 += xqs[192 + dd] * wv;
      }
      X2s[tid] = geluf(z0);
      X2s[256 + tid] = geluf(z1);
      X2s[512 + tid] = geluf(z2);
      X2s[768 + tid] = geluf(z3);
    }
    __syncthreads();
    {
      float z = b2s[n64];
      for (int kk = 0; kk < 256; kk++) z += X2s[m4 * 256 + kk] * W2s[kk * S2 + n64];
      Z2s[tid] = z;
    }
    __syncthreads();
    if (tid < 4) {
      float s = 0.f, ss = 0.f;
      for (int n = 0; n < 64; n++) {
        float z = Z2s[tid * 64 + n];
        s += z; ss += z * z;
      }
      float mu = s * (1.f / 64.f);
      red[tid * 4 + 0] = mu;
      red[tid * 4 + 1] = sqrtf(ss * (1.f / 64.f) - mu * mu + 1e-6f);
    }
    __syncthreads();
    {
      float zh = (Z2s[tid] - red[m4 * 4 + 0]) / red[m4 * 4 + 1];
      out[gro] = xqs[tid] + lnw[n64] * zh + lnb[n64];
    }
    __syncthreads();
  }
}

// ---------------------------------------------------------------------------
// Final 64 -> 1 projection.
// ---------------------------------------------------------------------------
__global__ void __launch_bounds__(256) fc3_kernel(
    const float* __restrict__ h, const float* __restrict__ w,
    const float* __restrict__ b, float* __restrict__ out, int M) {
  int r = blockIdx.x * blockDim.x + threadIdx.x;
  if (r < M) {
    float s = b[0];
    for (int c = 0; c < 64; c++) s += h[(size_t)r * 64 + c] * w[c];
    out[r] = s;
  }
}

// ---------------------------------------------------------------------------
extern "C" void kernel_launch(void* const* d_in, const int* in_sizes, int n_in,
                              void* d_out, int out_size, void* d_ws,
                              size_t ws_size, hipStream_t stream) {
  (void)in_sizes; (void)n_in; (void)out_size; (void)ws_size;
  const float* x = (const float*)d_in[0];
  const float* conv1_w = (const float*)d_in[1];
  const float* conv1_b = (const float*)d_in[2];
  const float* convs_w = (const float*)d_in[3];
  const float* convs_b = (const float*)d_in[4];
  const float* ln_w = (const float*)d_in[5];
  const float* ln_b = (const float*)d_in[6];
  const float* wq = (const float*)d_in[7];
  const float* wk = (const float*)d_in[8];
  const float* wv = (const float*)d_in[9];
  const float* wo = (const float*)d_in[10];
  const float* ttt_W1 = (const float*)d_in[11];
  const float* ttt_b1 = (const float*)d_in[12];
  const float* ttt_W2 = (const float*)d_in[13];
  const float* ttt_b2 = (const float*)d_in[14];
  const float* ttt_lnw = (const float*)d_in[15];
  const float* ttt_lnb = (const float*)d_in[16];
  const float* fc_w = (const float*)d_in[17];
  const float* fc_b = (const float*)d_in[18];
  const float* fc2_w = (const float*)d_in[19];
  const float* fc2_b = (const float*)d_in[20];
  const float* fc3_w = (const float*)d_in[21];
  const float* fc3_b = (const float*)d_in[22];

  const int M = 4096;  // B*L
  float* ws = (float*)d_ws;
  float* hA = ws;                 // 4096x256
  float* hB = hA + 4096 * 256;    // 4096x256
  float* hsum = hB + 4096 * 256;  // 4096x256
  float* qb = hsum + 4096 * 256;  // 4096x256
  float* kb = qb + 4096 * 256;    // 4096x256
  float* vb = kb + 4096 * 256;    // 4096x256
  float* tb = vb + 4096 * 256;    // 4096x256 (ttt output)
  float* h2 = tb + 4096 * 256;    // 4096x256
  float* f1 = h2 + 4096 * 256;    // 4096x512
  float* f2 = f1 + 4096 * 512;    // 4096x64

  dim3 blk(256);

  // conv1 (1x1, weights (out,in)) + ReLU -> hA   [units = 256*4 = 1024]
  gemm_wmma<true, true, true, false><<<128, blk, 0, stream>>>(
      x, conv1_w, conv1_b, nullptr, hA, M, 256, 64);

  // 8 conv3 layers, ping-pong, skip-sum of layers 1,3,5,7 into hsum
  float* cur = hA;
  float* nxt = hB;
  for (int i = 0; i < 8; i++) {
    int sumMode = (i & 1) ? ((i == 1) ? 1 : 2) : 0;
    conv3_wmma<<<128, blk, 0, stream>>>(cur, convs_w + (size_t)i * 256 * 256 * 3,
                                        convs_b + i * 256, nxt, hsum, sumMode);
    float* t = cur; cur = nxt; nxt = t;
  }

  // LN -> hA (input to TTT, also the residual base)
  layernorm256<<<4096, blk, 0, stream>>>(hsum, ln_w, ln_b, hA);

  // q/k/v projections (layout (B,L,H*d) == per-head slices at h*64)
  gemm_wmma<false, false, false, false><<<128, blk, 0, stream>>>(
      hA, wq, nullptr, nullptr, qb, M, 256, 256);
  gemm_wmma<false, false, false, false><<<128, blk, 0, stream>>>(
      hA, wk, nullptr, nullptr, kb, M, 256, 256);
  gemm_wmma<false, false, false, false><<<128, blk, 0, stream>>>(
      hA, wv, nullptr, nullptr, vb, M, 256, 256);

  // TTT scan: 16 chains, LDS-resident fast weights
  size_t smemFloats = 64 * S1 + 256 * S2 + 256 + 64 + 64 + 64 + 3 * 256 +
                      3 * 1024 + 3 * 256 + 16;
  ttt_scan<<<16, blk, smemFloats * sizeof(float), stream>>>(
      qb, kb, vb, ttt_W1, ttt_b1, ttt_W2, ttt_b2, ttt_lnw, ttt_lnb, tb);

  // output projection + residual -> h2
  gemm_wmma<false, false, false, true><<<128, blk, 0, stream>>>(
      tb, wo, nullptr, hA, h2, M, 256, 256);

  // LN -> hB
  layernorm256<<<4096, blk, 0, stream>>>(h2, ln_w, ln_b, hB);

  // fc chain   [fc1 units = 256*8 = 2048 -> 256 blocks; fc2 units = 256 -> 32]
  gemm_wmma<false, true, false, false><<<256, blk, 0, stream>>>(
      hB, fc_w, fc_b, nullptr, f1, M, 512, 256);
  gemm_wmma<false, true, false, false><<<32, blk, 0, stream>>>(
      f1, fc2_w, fc2_b, nullptr, f2, M, 64, 512);
  fc3_kernel<<<16, blk, 0, stream>>>(f2, fc3_w, fc3_b, (float*)d_out, M);
}